// GNN_81733227643007
// MI455X (gfx1250) — compile-verified
//
#include <hip/hip_runtime.h>

typedef float v2f __attribute__((ext_vector_type(2)));
typedef float v8f __attribute__((ext_vector_type(8)));
typedef unsigned int u32x4 __attribute__((ext_vector_type(4)));
typedef int i32x4 __attribute__((ext_vector_type(4)));
typedef int i32x8 __attribute__((ext_vector_type(8)));

// GEMM1 tiling
#define ROWS_PER_BLOCK 128   // 8 waves x 16 rows
#define KC 128               // K chunk staged per TDM load
#define ROW_PAD 4            // pad 4 dwords per 128 dwords (TDM pad feature)
#define ROW_STRIDE (KC + ROW_PAD)            // 132 floats
#define BUF_FLOATS (ROWS_PER_BLOCK * ROW_STRIDE)
#define KSPLIT 8

// ---------------------------------------------------------------------------
// Degree / normalization
// ---------------------------------------------------------------------------
__global__ void k_init_deg(float* __restrict__ deg, int n) {
    int i = blockIdx.x * blockDim.x + threadIdx.x;
    if (i < n) deg[i] = 1.0f;  // self-loop contributes 1 to every node's degree
}

__global__ void k_deg_edges(const int* __restrict__ dst, float* __restrict__ deg, int E) {
    int e = blockIdx.x * blockDim.x + threadIdx.x;
    if (e < E) atomicAdd(&deg[dst[e]], 1.0f);
}

__global__ void k_rsqrt(float* __restrict__ deg, int n) {
    int i = blockIdx.x * blockDim.x + threadIdx.x;
    if (i < n) deg[i] = rsqrtf(deg[i]);  // deg >= 1 always (self-loops)
}

__global__ void k_zero(float* __restrict__ p, int cnt) {
    int i = blockIdx.x * blockDim.x + threadIdx.x;
    if (i < cnt) p[i] = 0.0f;
}

// ---------------------------------------------------------------------------
// TDM: issue tensor_load_to_lds of a 128-row x 128-col fp32 tile of x
// (row stride n floats) into LDS at ldsByteAddr, with 4-dword padding after
// every 128 dwords so each LDS row is 132 floats (bank-conflict-free reads).
// Descriptor built per CDNA5 D# group0/group1 bit layout.
// ---------------------------------------------------------------------------
__device__ __forceinline__ void tdm_load_tile(const float* gsrc,
                                              unsigned ldsByteAddr, int n) {
    unsigned long long ga = (unsigned long long)(uintptr_t)gsrc;

    u32x4 g0;
    g0[0] = 1u;                                  // count=1, user descriptor
    g0[1] = ldsByteAddr;                         // lds_addr (bytes)
    g0[2] = (unsigned)(ga & 0xFFFFFFFFu);        // global_addr[31:0]
    g0[3] = ((unsigned)(ga >> 32) & 0x1FFFFFFu)  // global_addr[56:32]
          | (2u << 30);                          // type = 2 ("image")

    i32x8 g1;
    g1[0] = (int)((2u << 16)       // data_size = 4 bytes
                | (1u << 20)       // pad_enable
                | (6u << 22)       // pad_interval: 128 dwords
                | (3u << 25));     // pad_amount:  4 dwords
    g1[1] = (int)(((unsigned)n & 0xFFFFu) << 16);          // tensor_dim0 lo16
    g1[2] = (int)((((unsigned)n >> 16) & 0xFFFFu)          // tensor_dim0 hi16
                | (((unsigned)n & 0xFFFFu) << 16));        // tensor_dim1 lo16
    g1[3] = (int)((((unsigned)n >> 16) & 0xFFFFu)          // tensor_dim1 hi16
                | ((unsigned)KC << 16));                   // tile_dim0 = 128
    g1[4] = (int)(unsigned)ROWS_PER_BLOCK;                 // tile_dim1 = 128
    g1[5] = n;                                             // dim0_stride lo32
    g1[6] = 0;
    g1[7] = 0;

    i32x4 gz4 = {0, 0, 0, 0};
    i32x8 gz8 = {0, 0, 0, 0, 0, 0, 0, 0};
    __builtin_amdgcn_tensor_load_to_lds(g0, g1, gz4, gz4, gz8, 0);
}

// ---------------------------------------------------------------------------
// GEMM1: h1[N,16] += x[rows,Kseg] @ W1[Kseg,16] via V_WMMA_F32_16X16X4_F32.
// 8 waves/block, each wave owns a 16-row tile; K split across KSPLIT block
// groups for HBM-saturating concurrency; partials merged via f32 atomics.
// A-tiles staged to LDS by the Tensor Data Mover, double-buffered.
// ---------------------------------------------------------------------------
__global__ void k_gemm1_wmma(const float* __restrict__ x,
                             const float* __restrict__ W1,
                             float* __restrict__ h1,
                             int n, int rowTiles, int kseg) {
    extern __shared__ float smem[];

    const int waveId = threadIdx.x >> 5;
    const int lane   = threadIdx.x & 31;
    const int m      = lane & 15;   // M row within 16-row tile (N col for B/C)
    const int hh     = lane >> 4;   // half-wave: K offset 0 or 2

    const int rt   = blockIdx.x % rowTiles;
    const int ks   = blockIdx.x / rowTiles;
    const int row0 = rt * ROWS_PER_BLOCK;
    const int k0   = ks * kseg;
    const int S    = kseg / KC;     // stages

    const unsigned ldsBase = (unsigned)(uintptr_t)smem;
    const float* gtile0 = x + (size_t)row0 * (size_t)n + k0;

    // per-lane LDS read base: row r = waveId*16 + m, col offset 2*hh
    const int r = waveId * 16 + m;
    const float* __restrict__ ap0 = smem + r * ROW_STRIDE + 2 * hh;

    // per-lane W1 base: element (k)*16 + m with half-wave K offset (+2)*16
    const float* __restrict__ wp = W1 + 32 * hh + m;

    if (waveId == 0) tdm_load_tile(gtile0, ldsBase, n);

    v8f c = {};
    for (int s = 0; s < S; ++s) {
        if (waveId == 0) {
            if (s + 1 < S) {
                tdm_load_tile(gtile0 + (size_t)(s + 1) * KC,
                              ldsBase + ((s + 1) & 1) * (BUF_FLOATS * 4), n);
                // oldest TDM load (stage s) done when <=1 remains in flight
                __builtin_amdgcn_s_wait_tensorcnt(1);
            } else {
                __builtin_amdgcn_s_wait_tensorcnt(0);
            }
        }
        __syncthreads();

        const float* __restrict__ ap  = ap0 + (s & 1) * BUF_FLOATS;
        const float* __restrict__ wpk = wp + (size_t)(k0 + s * KC) * 16;
#pragma unroll
        for (int kk = 0; kk < KC; kk += 4) {
            v2f a = *(const v2f*)(ap + kk);        // ds_load_b64, conflict-free
            v2f b;
            b.x = wpk[kk * 16];                    // W1[(k+kk+2hh)*16 + m]
            b.y = wpk[kk * 16 + 16];               // W1[(k+kk+1+2hh)*16 + m]
            c = __builtin_amdgcn_wmma_f32_16x16x4_f32(
                false, a, false, b, (short)0, c, false, false);
        }
        __syncthreads();
    }

    // C/D layout: VGPR v, lanes 0-15 -> M=v ; lanes 16-31 -> M=v+8 ; N=m
    const int rowBase = row0 + waveId * 16;
#pragma unroll
    for (int v = 0; v < 8; ++v) {
        atomicAdd(&h1[(size_t)(rowBase + v + 8 * hh) * 16 + m], c[v]);
    }
}

// ---------------------------------------------------------------------------
// Aggregation: agg[d] = b + sum_e norm(e) * h[src(e)]  (+ self-loop term)
// ---------------------------------------------------------------------------
template <int F>
__global__ void k_init_agg(const float* __restrict__ h,
                           const float* __restrict__ dis,
                           const float* __restrict__ bias,
                           float* __restrict__ agg, int n) {
    int i = blockIdx.x * blockDim.x + threadIdx.x;
    if (i >= n * F) return;
    int node = i / F;
    int j    = i - node * F;
    float d  = dis[node];
    agg[i] = bias[j] + h[i] * d * d;  // self-loop: norm = dis[i]*dis[i]
}

template <int F>
__global__ void k_edge_agg(const int* __restrict__ src,
                           const int* __restrict__ dst,
                           const float* __restrict__ dis,
                           const float* __restrict__ h,
                           float* __restrict__ agg, int E) {
    int t = blockIdx.x * blockDim.x + threadIdx.x;
    int e = t / F;
    int j = t - e * F;
    if (e >= E) return;
    int s = src[e], d = dst[e];
    float w = dis[s] * dis[d];
    atomicAdd(&agg[(size_t)d * F + j], h[(size_t)s * F + j] * w);
}

// ---------------------------------------------------------------------------
// GEMM2 with fused ReLU on the input: h2[N,3] = relu(agg1) @ W2[16,3]
// ---------------------------------------------------------------------------
__global__ void k_gemm2_relu(const float* __restrict__ agg1,
                             const float* __restrict__ W2,
                             float* __restrict__ h2, int n) {
    int i = blockIdx.x * blockDim.x + threadIdx.x;
    if (i >= n) return;
    float a0 = 0.f, a1 = 0.f, a2 = 0.f;
#pragma unroll
    for (int k = 0; k < 16; ++k) {
        float v = fmaxf(agg1[(size_t)i * 16 + k], 0.0f);
        a0 = fmaf(v, W2[k * 3 + 0], a0);
        a1 = fmaf(v, W2[k * 3 + 1], a1);
        a2 = fmaf(v, W2[k * 3 + 2], a2);
    }
    h2[(size_t)i * 3 + 0] = a0;
    h2[(size_t)i * 3 + 1] = a1;
    h2[(size_t)i * 3 + 2] = a2;
}

// ---------------------------------------------------------------------------
// Row softmax over 3 classes
// ---------------------------------------------------------------------------
__global__ void k_softmax3(const float* __restrict__ agg2,
                           float* __restrict__ out, int n) {
    int i = blockIdx.x * blockDim.x + threadIdx.x;
    if (i >= n) return;
    float v0 = agg2[(size_t)i * 3 + 0];
    float v1 = agg2[(size_t)i * 3 + 1];
    float v2 = agg2[(size_t)i * 3 + 2];
    float mx = fmaxf(v0, fmaxf(v1, v2));
    float e0 = __expf(v0 - mx), e1 = __expf(v1 - mx), e2 = __expf(v2 - mx);
    float inv = 1.0f / (e0 + e1 + e2);
    out[(size_t)i * 3 + 0] = e0 * inv;
    out[(size_t)i * 3 + 1] = e1 * inv;
    out[(size_t)i * 3 + 2] = e2 * inv;
}

// ---------------------------------------------------------------------------
extern "C" void kernel_launch(void* const* d_in, const int* in_sizes, int n_in,
                              void* d_out, int out_size, void* d_ws, size_t ws_size,
                              hipStream_t stream) {
    const float* x   = (const float*)d_in[0];
    const int*   ei  = (const int*)d_in[1];
    const float* W1  = (const float*)d_in[2];
    const float* b1  = (const float*)d_in[3];
    const float* W2  = (const float*)d_in[4];
    const float* b2  = (const float*)d_in[5];
    float*       out = (float*)d_out;

    const int H = in_sizes[3];           // 16
    const int n = in_sizes[2] / H;       // 16384
    const int E = in_sizes[1] / 2;       // 524288
    const int* src = ei;
    const int* dst = ei + E;

    // Workspace layout (floats): dis[n] | h1[16n] | agg1[16n] | h2[3n] | agg2[3n]
    float* ws   = (float*)d_ws;
    float* dis  = ws;
    float* h1   = ws + (size_t)n;
    float* agg1 = ws + (size_t)17 * n;
    float* h2   = ws + (size_t)33 * n;
    float* agg2 = ws + (size_t)36 * n;

    const int T = 256;

    // 1) symmetric normalization coefficients
    k_init_deg<<<(n + T - 1) / T, T, 0, stream>>>(dis, n);
    k_deg_edges<<<(E + T - 1) / T, T, 0, stream>>>(dst, dis, E);
    k_rsqrt<<<(n + T - 1) / T, T, 0, stream>>>(dis, n);

    // 2) layer 1 linear: h1 = x @ W1 (TDM-staged, K-split WMMA, f32 atomics)
    k_zero<<<(n * 16 + T - 1) / T, T, 0, stream>>>(h1, n * 16);
    {
        const int rowTiles = n / ROWS_PER_BLOCK;      // 128
        const int kseg     = n / KSPLIT;              // 2048
        const int grid     = rowTiles * KSPLIT;       // 1024 blocks, 8192 waves
        const size_t smemBytes = (size_t)2 * BUF_FLOATS * sizeof(float);
        k_gemm1_wmma<<<grid, 256, smemBytes, stream>>>(x, W1, h1, n, rowTiles, kseg);
    }

    // 3) layer 1 aggregation (+bias), fused self-loop in init
    k_init_agg<16><<<(n * 16 + T - 1) / T, T, 0, stream>>>(h1, dis, b1, agg1, n);
    k_edge_agg<16><<<(E * 16 + T - 1) / T, T, 0, stream>>>(src, dst, dis, h1, agg1, E);

    // 4) layer 2 linear with fused ReLU
    k_gemm2_relu<<<(n + T - 1) / T, T, 0, stream>>>(agg1, W2, h2, n);

    // 5) layer 2 aggregation (+bias)
    k_init_agg<3><<<(n * 3 + T - 1) / T, T, 0, stream>>>(h2, dis, b2, agg2, n);
    k_edge_agg<3><<<(E * 3 + T - 1) / T, T, 0, stream>>>(src, dst, dis, h2, agg2, E);

    // 6) softmax
    k_softmax3<<<(n + T - 1) / T, T, 0, stream>>>(agg2, out, n);
}